// LogisticRegressionClassifier_78726750536374
// MI455X (gfx1250) — compile-verified
//
#include <hip/hip_runtime.h>
#include <hip/hip_bf16.h>
#include <math.h>

#define VOCAB   512
#define MAXLEN  512
#define BATCH   128
#define NBINS   (VOCAB + VOCAB * VOCAB + 2)   // 262658
#define NTHREADS 256

typedef unsigned v4u __attribute__((ext_vector_type(4)));
typedef unsigned v8u __attribute__((ext_vector_type(8)));

// One block per batch row. CDNA5 data-movement paths exercised:
//  - tensor_load_to_lds (TDM, TENSORcnt)      : DMA unigram W (2KB) into LDS, issued by wave 0
//  - global_load_async_to_lds_b128 (ASYNCcnt) : stage x row (4KB) into LDS
//  - s_wait_tensorcnt / s_wait_asynccnt       : counter fences before the barrier
//  - global_prefetch_b8                       : warm L2 with bigram-table slice
__global__ __launch_bounds__(NTHREADS) void ngram_logreg_kernel(
    const float* __restrict__ x,        // [B, MAXLEN, 2] (action, time)
    const int*   __restrict__ lengths,  // [B]
    const float* __restrict__ W,        // [NBINS]
    const float* __restrict__ bias,     // [1]
    float* __restrict__ out)            // [B]
{
    __shared__ float sX[MAXLEN * 2];    // 4 KB: staged x row
    __shared__ float sW[VOCAB];         // 2 KB: unigram weights (TDM destination)
    __shared__ float sPart[NTHREADS / 32];
    __shared__ float sTmax[NTHREADS / 32];

    const int b   = blockIdx.x;
    const int tid = threadIdx.x;
    const float* xrow = x + (size_t)b * (MAXLEN * 2);

    // Warm L2 with this block's 8KB slice of the 1MB bigram table
    // (emits global_prefetch_b8; speculative, no counters).
    {
        const char* p = (const char*)(W + VOCAB) + (size_t)b * 8192 + (size_t)tid * 32;
        __builtin_prefetch(p, 0, 1);
    }

    // --- TDM: wave 0 DMAs the 2KB unigram table into LDS (TENSORcnt path) ---
    if (tid < 32) {
        const unsigned lds_w = (unsigned)(size_t)(&sW[0]);
        const unsigned long long ga = (unsigned long long)W;
        // D# group 0: count=1 | lds_addr[63:32] | global_addr[120:64] | type=2 [127:126]
        v4u g0;
        g0.x = 1u;                                    // count=1, user mode, no gather
        g0.y = lds_w;                                 // LDS byte address
        g0.z = (unsigned)ga;                          // global_addr[31:0]
        g0.w = ((unsigned)(ga >> 32) & 0x01FFFFFFu)   // global_addr[56:32]
               | (2u << 30);                          // type=2 ("image")
        // D# group 1: data_size=2 (4B), tensor 512x1, tile 512x1, stride0=512
        v8u g1;
        g1[0] = 2u << 16;       // [17:16] data_size = 4 bytes; mask/flags = 0
        g1[1] = 512u << 16;     // [63:48]  tensor_dim0 low16 = 512
        g1[2] = 1u << 16;       // [95:80]  tensor_dim1 low16 = 1
        g1[3] = 512u << 16;     // [127:112] tile_dim0 = 512
        g1[4] = 1u;             // [143:128] tile_dim1 = 1 (tile_dim2 = 0)
        g1[5] = 512u;           // [191:160] tensor_dim0_stride low32 = 512
        g1[6] = 0u;             // stride0 high / tensor_dim1_stride low = 0
        g1[7] = 0u;             // tensor_dim1_stride rest = 0
        asm volatile("tensor_load_to_lds %0, %1" :: "s"(g0), "s"(g1) : "memory");
        __builtin_amdgcn_s_wait_tensorcnt(0);
    }

    // --- async stage of the x row into LDS (ASYNCcnt path), 16B per thread ---
    {
        unsigned long long gx = (unsigned long long)(xrow + tid * 4);
        unsigned lds_x = (unsigned)(size_t)(&sX[tid * 4]);
        asm volatile("global_load_async_to_lds_b128 %0, %1, off"
                     :: "v"(lds_x), "v"(gx) : "memory");
        asm volatile("s_wait_asynccnt 0" ::: "memory");
    }
    __syncthreads();

    const int len = lengths[b];
    float acc  = 0.0f;
    float tmax = -INFINITY;

#pragma unroll
    for (int k = 0; k < MAXLEN / NTHREADS; ++k) {
        const int j = tid + k * NTHREADS;
        const int   a  = (int)sX[2 * j];
        const float tj = sX[2 * j + 1];
        if (j < len) {
            acc += sW[a];                       // unigram from LDS (TDM-staged)
            tmax = fmaxf(tmax, tj);
        }
        if (j + 1 < len) {                      // bi_w = valid[:, 1:]
            const int a2 = (int)sX[2 * j + 2];  // j+1 <= 511 when taken
            acc += W[VOCAB + a * VOCAB + a2];   // bigram gather (L2)
        }
    }

    // wave32 reduction: sum(acc), max(tmax)
#pragma unroll
    for (int off = 16; off > 0; off >>= 1) {
        acc  += __shfl_xor(acc, off, 32);
        tmax  = fmaxf(tmax, __shfl_xor(tmax, off, 32));
    }
    const int wave = tid >> 5;
    if ((tid & 31) == 0) { sPart[wave] = acc; sTmax[wave] = tmax; }
    __syncthreads();

    if (tid == 0) {
        float a = 0.0f, tm = -INFINITY;
#pragma unroll
        for (int w = 0; w < NTHREADS / 32; ++w) {
            a  += sPart[w];
            tm  = fmaxf(tm, sTmax[w]);
        }
        if (len <= 0) tm = 0.0f;   // reference guards lengths==0 (can't occur, lengths>=1)
        out[b] = a + W[NBINS - 2] * tm + W[NBINS - 1] * (float)len + bias[0];
    }
}

extern "C" void kernel_launch(void* const* d_in, const int* in_sizes, int n_in,
                              void* d_out, int out_size, void* d_ws, size_t ws_size,
                              hipStream_t stream) {
    const float* x       = (const float*)d_in[0];  // [B, MAXLEN, 2]
    const int*   lengths = (const int*)d_in[1];    // [B]
    const float* W       = (const float*)d_in[2];  // [NBINS]
    const float* bias    = (const float*)d_in[3];  // [1]
    float* out = (float*)d_out;                    // [B]

    (void)in_sizes; (void)n_in; (void)out_size; (void)d_ws; (void)ws_size;

    ngram_logreg_kernel<<<BATCH, NTHREADS, 0, stream>>>(x, lengths, W, bias, out);
}